// ExpFlow_41480794144774
// MI455X (gfx1250) — compile-verified
//
#include <hip/hip_runtime.h>

// ---------------------------------------------------------------------------
// expv (scaling-and-squaring of a 3D velocity field), MI455X / gfx1250.
//
// Shapes: x  [N=2, C=3, D=128, H=128, W=128] float32 (planar)
//         out same shape/type.
// u is kept in INTERLEAVED layout [N,D,H,W,3] (12 B/voxel) in d_ws so each of
// the 8 trilinear gathers is a single contiguous 12-byte load (b96 path).
// Two ping-pong buffers: 2 * 2*128^3 * 3 * 4B = 100.7 MB of d_ws.
// Both buffers fit in the 192 MB L2, so all 5 squaring steps are L2-resident.
// Step 5 is fused with the interleaved->planar transpose into d_out.
// ---------------------------------------------------------------------------

#define WBITS 7u
#define WDIM  128
#define HWSH  14u                  // log2(H*W)
#define DHWSH 21u                  // log2(D*H*W)
#define DHW   (1u << DHWSH)        // 2,097,152 voxels per batch item
#define NVOX  (2u * DHW)           // total voxels (N=2)
#define SCALE_INIT (1.0f / 32.0f)  // SCALE / 2^STEPS

struct V3 { float x, y, z; };

__device__ __forceinline__ V3 ld3(const float* __restrict__ p) {
    V3 v; v.x = p[0]; v.y = p[1]; v.z = p[2]; return v;
}
__device__ __forceinline__ V3 lerp3(V3 a, V3 b, float w) {
    V3 r;
    r.x = fmaf(w, b.x - a.x, a.x);
    r.y = fmaf(w, b.y - a.y, a.y);
    r.z = fmaf(w, b.z - a.z, a.z);
    return r;
}

// planar x [N,3,DHW] -> interleaved u [N*DHW, 3], scaled by 1/2^STEPS
__global__ __launch_bounds__(256) void expv_init(const float* __restrict__ x,
                                                 float* __restrict__ u) {
    unsigned v = blockIdx.x * 256u + threadIdx.x;
    if (v >= NVOX) return;
    unsigned n = v >> DHWSH;
    unsigned r = v & (DHW - 1u);
    const float* xb = x + (size_t)n * 3u * DHW + r;
    float* o = u + (size_t)v * 3u;
    o[0] = xb[0]        * SCALE_INIT;
    o[1] = xb[DHW]      * SCALE_INIT;
    o[2] = xb[2u * DHW] * SCALE_INIT;
}

// One squaring step: uout = uin + trilinear_sample(uin, id + uin).
// FINAL=true writes planar [N,3,DHW] (fused output transpose), else interleaved.
template <bool FINAL>
__global__ __launch_bounds__(256) void expv_step(const float* __restrict__ uin,
                                                 float* __restrict__ uout) {
    unsigned v = blockIdx.x * 256u + threadIdx.x;
    if (v >= NVOX) return;
    unsigned n  = v >> DHWSH;
    unsigned r  = v & (DHW - 1u);
    unsigned xi = r & 127u;
    unsigned yi = (r >> WBITS) & 127u;
    unsigned zi = r >> HWSH;

    V3 u = ld3(uin + (size_t)v * 3u);

    // align_corners identity grid + displacement, unnormalized to pixels:
    // pix = i + 0.5*(size-1)*u ; border padding == clamp before floor.
    float px = fminf(fmaxf((float)xi + 63.5f * u.x, 0.0f), 127.0f);
    float py = fminf(fmaxf((float)yi + 63.5f * u.y, 0.0f), 127.0f);
    float pz = fminf(fmaxf((float)zi + 63.5f * u.z, 0.0f), 127.0f);

    float x0f = floorf(px), y0f = floorf(py), z0f = floorf(pz);
    int x0 = (int)x0f, y0 = (int)y0f, z0 = (int)z0f;
    int x1 = min(x0 + 1, 127), y1 = min(y0 + 1, 127), z1 = min(z0 + 1, 127);
    float wx = px - x0f, wy = py - y0f, wz = pz - z0f;

    const float* base = uin + ((size_t)n << DHWSH) * 3u;
    #define IDX(zz, yy, xx) (3u * (unsigned)(((zz) << HWSH) | ((yy) << WBITS) | (xx)))

    // cover gather latency on the rows issued last (gfx1250: global_prefetch_b8)
    __builtin_prefetch(base + IDX(z1, y0, x0), 0, 3);
    __builtin_prefetch(base + IDX(z1, y1, x0), 0, 3);

    V3 c000 = ld3(base + IDX(z0, y0, x0));
    V3 c001 = ld3(base + IDX(z0, y0, x1));
    V3 c010 = ld3(base + IDX(z0, y1, x0));
    V3 c011 = ld3(base + IDX(z0, y1, x1));
    V3 c100 = ld3(base + IDX(z1, y0, x0));
    V3 c101 = ld3(base + IDX(z1, y0, x1));
    V3 c110 = ld3(base + IDX(z1, y1, x0));
    V3 c111 = ld3(base + IDX(z1, y1, x1));
    #undef IDX

    V3 c00 = lerp3(c000, c001, wx);
    V3 c01 = lerp3(c010, c011, wx);
    V3 c10 = lerp3(c100, c101, wx);
    V3 c11 = lerp3(c110, c111, wx);
    V3 c0  = lerp3(c00, c01, wy);
    V3 c1  = lerp3(c10, c11, wy);
    V3 s   = lerp3(c0, c1, wz);

    V3 o; o.x = u.x + s.x; o.y = u.y + s.y; o.z = u.z + s.z;

    if (FINAL) {
        float* ob = uout + (size_t)n * 3u * DHW + r;   // planar [N,3,DHW]
        ob[0]        = o.x;
        ob[DHW]      = o.y;
        ob[2u * DHW] = o.z;
    } else {
        float* ob = uout + (size_t)v * 3u;             // interleaved
        ob[0] = o.x; ob[1] = o.y; ob[2] = o.z;
    }
}

extern "C" void kernel_launch(void* const* d_in, const int* in_sizes, int n_in,
                              void* d_out, int out_size, void* d_ws, size_t ws_size,
                              hipStream_t stream) {
    (void)in_sizes; (void)n_in; (void)out_size; (void)ws_size;
    const float* x = (const float*)d_in[0];
    float* out  = (float*)d_out;
    float* bufA = (float*)d_ws;                      // NVOX*3 floats = 50.3 MB
    float* bufB = bufA + (size_t)NVOX * 3u;          // total d_ws use: 100.7 MB

    dim3 block(256);
    dim3 grid(NVOX / 256u);                          // 16384 blocks, 8 wave32 each

    expv_init<<<grid, block, 0, stream>>>(x, bufA);
    expv_step<false><<<grid, block, 0, stream>>>(bufA, bufB);  // step 1
    expv_step<false><<<grid, block, 0, stream>>>(bufB, bufA);  // step 2
    expv_step<false><<<grid, block, 0, stream>>>(bufA, bufB);  // step 3
    expv_step<false><<<grid, block, 0, stream>>>(bufB, bufA);  // step 4
    expv_step<true ><<<grid, block, 0, stream>>>(bufA, out);   // step 5 + transpose
}